// VIN_Block_32590211842169
// MI455X (gfx1250) — compile-verified
//
#include <hip/hip_runtime.h>
#include <hip/hip_bf16.h>
#include <float.h>

// CDNA5 / gfx1250 VIN block.
//  - f32 WMMA 16x16x4 for the 10ch x 9tap conv-as-GEMM (M=chan pad16, K=taps pad12, N=16 pixels)
//  - one workgroup per batch image, all state in LDS, double-buffered value map, 1 barrier/iter.
//  - loop-invariant accumulator (qr) cached in VGPRs across the 39 iterations.

typedef __attribute__((ext_vector_type(2))) float v2f;
typedef __attribute__((ext_vector_type(8))) float v8f;

// Guard only in the device pass: __has_builtin() is false for amdgcn builtins in the host pass.
#if defined(__HIP_DEVICE_COMPILE__)
#if !__has_builtin(__builtin_amdgcn_wmma_f32_16x16x4_f32)
#error "wmma_f32_16x16x4_f32 builtin not available on this toolchain (device pass)"
#endif
#endif

#define HW     32
#define NPIX   1024
#define PW     34          // padded width (zero border for SAME conv)
#define PAD_N  (PW * PW)   // 1156
#define CH_Q   10
#define K_ITER 40
#define SBS    10

// One 16-pixel tile: D = A(16x12,padded from 10x9) * B(12x16 im2col) + C, via 3x K=4 f32 WMMAs.
// A layout (ISA 16x4 f32): lane&15 = M row; lanes>=16 supply taps K+2,K+3. B mirrored:
// lanes 0-15 rows K,K+1 ; lanes 16-31 rows K+2,K+3 ; column N = lane&15.
__device__ __forceinline__ v8f conv9_wmma(const float* vpad, int y, int x, int lo,
                                          v2f a0, v2f a1, v2f a2, v8f c)
{
    const float* base = vpad + y * PW + x;        // pad coords of neighbor (y-1,x-1)
    float n0 = base[0],      n1 = base[1],          n2 = base[2];
    float n3 = base[PW],     n4 = base[PW + 1],     n5 = base[PW + 2];
    float n6 = base[2 * PW], n7 = base[2 * PW + 1], n8 = base[2 * PW + 2];
    v2f b0, b1, b2;
    b0.x = lo ? n0 : n2;   b0.y = lo ? n1 : n3;   // taps 0,1 | 2,3
    b1.x = lo ? n4 : n6;   b1.y = lo ? n5 : n7;   // taps 4,5 | 6,7
    b2.x = lo ? n8 : 0.f;  b2.y = 0.f;            // taps 8,- | -,-
    c = __builtin_amdgcn_wmma_f32_16x16x4_f32(false, a0, false, b0, (short)0, c, false, false);
    c = __builtin_amdgcn_wmma_f32_16x16x4_f32(false, a1, false, b1, (short)0, c, false, false);
    c = __builtin_amdgcn_wmma_f32_16x16x4_f32(false, a2, false, b2, (short)0, c, false, false);
    return c;
}

__global__ __launch_bounds__(256) void vin_block_kernel(
    const float* __restrict__ X,  const int* __restrict__ S1, const int* __restrict__ S2,
    const float* __restrict__ bias, const float* __restrict__ w0, const float* __restrict__ w1,
    const float* __restrict__ w,  const float* __restrict__ wfb, const float* __restrict__ wo,
    float* __restrict__ out, float* __restrict__ pred)
{
    __shared__ float s_weff[18];
    __shared__ float s_beff;
    __shared__ float s_w[90];
    __shared__ float s_wfb[90];
    __shared__ float s_wo[80];
    __shared__ float s_rpad[PAD_N];
    __shared__ float s_vpad[2][PAD_N];
    __shared__ float s_qr[CH_Q * NPIX];          // first 2*PAD_N floats reused as padded X

    const int tid  = threadIdx.x;
    const int b    = blockIdx.x;
    const int lane = tid & 31;
    const int wave = tid >> 5;
    const int lo   = (lane < 16) ? 1 : 0;
    const int hi   = lane >> 4;
    const int col  = lane & 15;                  // N column / A row

    // ---- phase 0: prefetch X, zero padded arrays, stage weights, fold conv0+conv1 ----
    const float* Xb = X + (size_t)b * 2 * NPIX;
    __builtin_prefetch(Xb + tid * 8, 0, 0);      // global_prefetch_b8: cover 2048 floats w/ 256 lanes
    float* s_xpad = s_qr;                        // alias (2*PAD_N <= CH_Q*NPIX)
    for (int i = tid; i < 2 * PAD_N; i += 256) s_xpad[i] = 0.f;
    for (int i = tid; i < PAD_N; i += 256) { s_rpad[i] = 0.f; s_vpad[0][i] = 0.f; s_vpad[1][i] = 0.f; }
    for (int i = tid; i < 90; i += 256) s_w[i]   = w[i];
    for (int i = tid; i < 90; i += 256) s_wfb[i] = wfb[i];
    for (int i = tid; i < 80; i += 256) s_wo[i]  = wo[i];
    if (tid < 19) {                              // w_eff[i*9+t] = sum_c w1[c]*w0[c][i][t]
        float acc = 0.f;
        for (int c = 0; c < 150; ++c) {
            float s = w1[c];
            acc += (tid < 18) ? s * w0[c * 18 + tid] : s * bias[c];
        }
        if (tid < 18) s_weff[tid] = acc; else s_beff = acc;
    }
    __syncthreads();

    // ---- phase 1: stage X (zero-padded) ----
    for (int i = tid; i < 2 * NPIX; i += 256) {
        int ci = i >> 10, p = i & 1023;
        int y = p >> 5, x = p & 31;
        s_xpad[ci * PAD_N + (y + 1) * PW + (x + 1)] = Xb[i];
    }
    __syncthreads();

    // ---- phase 2: r = conv3x3(X, w_eff) + b_eff ----
    for (int p = tid; p < NPIX; p += 256) {
        int y = p >> 5, x = p & 31;
        float acc = s_beff;
        #pragma unroll
        for (int ci = 0; ci < 2; ++ci) {
            const float* bp = s_xpad + ci * PAD_N + y * PW + x;
            const float* wp = s_weff + ci * 9;
            acc += wp[0]*bp[0]      + wp[1]*bp[1]        + wp[2]*bp[2]
                 + wp[3]*bp[PW]     + wp[4]*bp[PW+1]     + wp[5]*bp[PW+2]
                 + wp[6]*bp[2*PW]   + wp[7]*bp[2*PW+1]   + wp[8]*bp[2*PW+2];
        }
        s_rpad[(y + 1) * PW + (x + 1)] = acc;
    }
    __syncthreads();

    // ---- per-lane A-matrix registers for w (qr pass) and w_fb (iteration pass) ----
    v2f aw0, aw1, aw2, af0, af1, af2;
    {
        const float* rw = s_w   + ((col < CH_Q) ? col : 0) * 9;
        const float* rf = s_wfb + ((col < CH_Q) ? col : 0) * 9;
        float tw[12], tf[12];
        #pragma unroll
        for (int i = 0; i < 12; ++i) {
            bool v = (col < CH_Q) && (i < 9);
            tw[i] = v ? rw[i] : 0.f;
            tf[i] = v ? rf[i] : 0.f;
        }
        int k = 2 * hi;
        aw0.x = tw[k];     aw0.y = tw[k + 1];
        aw1.x = tw[4 + k]; aw1.y = tw[5 + k];
        aw2.x = tw[8 + k]; aw2.y = tw[9 + k];
        af0.x = tf[k];     af0.y = tf[k + 1];
        af1.x = tf[4 + k]; af1.y = tf[5 + k];
        af2.x = tf[8 + k]; af2.y = tf[9 + k];
    }

    // ---- phase 3: qr[c] = conv3x3(r, w[c]) (loop-invariant); cache in VGPRs; v0 = max_c qr ----
    v8f cq[8];                                   // per-tile loop-invariant accumulator init
    #pragma unroll 1
    for (int t = 0; t < 8; ++t) {
        int p = (wave * 8 + t) * 16 + col;
        int y = p >> 5, x = p & 31;
        v8f c = {};
        c = conv9_wmma(s_rpad, y, x, lo, aw0, aw1, aw2, c);
        float m = -1e30f;
        #pragma unroll
        for (int j = 0; j < 8; ++j) {
            int cj = j + hi * 8;                 // C/D layout: VGPR j -> M = j (+8 for hi lanes)
            if (cj < CH_Q) { s_qr[cj * NPIX + p] = c[j]; m = fmaxf(m, c[j]); }
            cq[t][j] = (cj < CH_Q) ? c[j] : -1e30f;
        }
        m = fmaxf(m, __shfl_xor(m, 16));
        s_vpad[0][(y + 1) * PW + (x + 1)] = m;   // both half-lanes hold same m: store unconditionally
    }
    __syncthreads();

    // ---- phase 4: 39 value-iteration steps, double-buffered v, qr kept in registers ----
    #pragma unroll 1
    for (int it = 0; it < K_ITER - 1; ++it) {
        const float* vin = s_vpad[it & 1];
        float* vout = s_vpad[(it + 1) & 1];
        #pragma unroll 1
        for (int t = 0; t < 8; ++t) {
            int p = (wave * 8 + t) * 16 + col;
            int y = p >> 5, x = p & 31;
            v8f c = conv9_wmma(vin, y, x, lo, af0, af1, af2, cq[t]);
            float m = c[0];
            #pragma unroll
            for (int j = 1; j < 8; ++j) m = fmaxf(m, c[j]);
            m = fmaxf(m, __shfl_xor(m, 16));
            vout[(y + 1) * PW + (x + 1)] = m;    // both half-lanes identical: no exec branch
        }
        __syncthreads();
    }

    // ---- phase 5: sample 10 points, final q = qr + conv(v,wfb), 10->8 matmul, softmax ----
    if (tid < SBS) {
        const float* vfin = s_vpad[(K_ITER - 1) & 1];
        int ys = S1[b * SBS + tid];
        int xs = S2[b * SBS + tid];
        const float* bp = vfin + ys * PW + xs;
        float q[CH_Q];
        #pragma unroll
        for (int c = 0; c < CH_Q; ++c) {
            const float* wf = s_wfb + c * 9;
            float acc = s_qr[c * NPIX + ys * HW + xs];
            acc += wf[0]*bp[0]    + wf[1]*bp[1]      + wf[2]*bp[2]
                 + wf[3]*bp[PW]   + wf[4]*bp[PW+1]   + wf[5]*bp[PW+2]
                 + wf[6]*bp[2*PW] + wf[7]*bp[2*PW+1] + wf[8]*bp[2*PW+2];
            q[c] = acc;
        }
        float logits[8], mx = -1e30f;
        #pragma unroll
        for (int o = 0; o < 8; ++o) {
            float acc = 0.f;
            #pragma unroll
            for (int c = 0; c < CH_Q; ++c) acc += q[c] * s_wo[o * 10 + c];
            logits[o] = acc;
            mx = fmaxf(mx, acc);
        }
        float e[8], se = 0.f;
        #pragma unroll
        for (int o = 0; o < 8; ++o) { e[o] = __expf(logits[o] - mx); se += e[o]; }
        float inv = 1.f / se;
        float* op = out  + (size_t)(b * SBS + tid) * 8;
        float* pp = pred + (size_t)(b * SBS + tid) * 8;
        #pragma unroll
        for (int o = 0; o < 8; ++o) { op[o] = logits[o]; pp[o] = e[o] * inv; }
    }
}

extern "C" void kernel_launch(void* const* d_in, const int* in_sizes, int n_in,
                              void* d_out, int out_size, void* d_ws, size_t ws_size,
                              hipStream_t stream) {
    const float* X    = (const float*)d_in[0];
    const int*   S1   = (const int*)d_in[1];
    const int*   S2   = (const int*)d_in[2];
    const float* bias = (const float*)d_in[3];
    const float* w0   = (const float*)d_in[4];
    const float* w1   = (const float*)d_in[5];
    const float* w    = (const float*)d_in[6];
    const float* wfb  = (const float*)d_in[7];
    const float* wo   = (const float*)d_in[8];
    int bs = in_sizes[0] / (2 * HW * HW);        // 256
    float* out  = (float*)d_out;
    float* pred = out + (size_t)bs * SBS * 8;
    vin_block_kernel<<<bs, 256, 0, stream>>>(X, S1, S2, bias, w0, w1, w, wfb, wo, out, pred);
}